// expansionPenaltyModule_30133490549120
// MI455X (gfx1250) — compile-verified
//
#include <hip/hip_runtime.h>
#include <math.h>

#define PRIM_M 512
#define NWAVES (PRIM_M / 32)
#define EPS_F 1e-12f
#define INF_SQ 1e20f   // squared-space sentinel (reference masks with 1e10 on lengths)

typedef float v2f __attribute__((ext_vector_type(2)));
typedef float v8f __attribute__((ext_vector_type(8)));

__device__ __forceinline__ unsigned long long pack_slot(float d, int p) {
    return ((unsigned long long)__float_as_uint(d) << 32) | (unsigned int)p;
}

// One workgroup per primitive (512 points). 512 threads = 16 wave32s.
// Points staged via CDNA5 async global->LDS DMA (ASYNCcnt). Prim's MST on
// squared distances (monotone-equivalent to the reference), packed 64-bit
// (dist,parent) LDS slots with visited folded into the sign, wave32 shuffle
// argmin, final edge norms via V_WMMA_F32_16X16X4_F32 Gram diagonal.
__global__ __launch_bounds__(PRIM_M) void mst_expansion_kernel(
    const float* __restrict__ xyz,
    const float* __restrict__ alpha_ptr,
    float* __restrict__ dist_out,
    int*   __restrict__ assign_out,
    float* __restrict__ mean_ws,
    int P)
{
    __shared__ float4             pts[PRIM_M];
    __shared__ unsigned long long slot[PRIM_M]; // {best_d_sq bits:32 | best_p:32}; hi = -1.0f => visited
    __shared__ int                parent_s[PRIM_M];
    __shared__ float              cost_s[PRIM_M];
    __shared__ float              wd[NWAVES];
    __shared__ int                wj[NWAVES];
    __shared__ float              wsum[NWAVES];
    __shared__ int                sel_u;
    __shared__ float              mean_sh;

    const int g    = blockIdx.x;      // primitive id in [0, B*P)
    const int tid  = threadIdx.x;     // node id this thread owns
    const int lane = tid & 31;
    const int wave = tid >> 5;

    // ---- stage points into LDS via async global->LDS DMA ----
    // ISA: the instruction offset is added to both the LDS and the global address,
    // so one (lds_base, gaddr) pair + offset:0/4/8 maps the 12B point into
    // the 16B-stride float4 slot. Tracked by ASYNCcnt (not LOADcnt).
    {
        const float*   gaddr   = xyz + (size_t)g * (PRIM_M * 3) + (size_t)tid * 3;
        const unsigned lds_off = (unsigned)(size_t)&pts[tid]; // low 32 bits = LDS byte offset
        asm volatile(
            "global_load_async_to_lds_b32 %0, %1, off\n\t"
            "global_load_async_to_lds_b32 %0, %1, off offset:4\n\t"
            "global_load_async_to_lds_b32 %0, %1, off offset:8"
            :
            : "v"(lds_off), "v"(gaddr)
            : "memory");
        asm volatile("s_wait_asynccnt 0" ::: "memory"); // this wave's DMA done
    }
    __syncthreads();                                    // publish LDS to all waves

    const float4 me = pts[tid];
    const float px = me.x, py = me.y, pz = me.z;

    // ---- init: best_d_sq = clamped squared D[0] row; node 0 visited (-1 marker) ----
    {
        const float4 p0 = pts[0];
        const float dx = px - p0.x, dy = py - p0.y, dz = pz - p0.z;
        const float d0sq = fmaxf(dx * dx + dy * dy + dz * dz, EPS_F);
        slot[tid] = pack_slot((tid == 0) ? -1.0f : d0sq, 0);
    }
    parent_s[tid] = -1;
    // (first loop barrier orders these inits vs. cross-thread reads)

    // ---- Prim's algorithm: 511 serial iterations ----
    for (int it = 0; it < PRIM_M - 1; ++it) {
        // candidate = where(visited, INF, best_d_sq); argmin, lowest-index tie-break
        const unsigned long long s = slot[tid];
        const float bd = __uint_as_float((unsigned int)(s >> 32));
        float d = (bd < 0.0f) ? INF_SQ : bd;
        int   j = tid;
        #pragma unroll
        for (int m = 16; m >= 1; m >>= 1) {
            const float od = __shfl_xor(d, m, 32);
            const int   oj = __shfl_xor(j, m, 32);
            if (od < d || (od == d && oj < j)) { d = od; j = oj; }
        }
        if (lane == 0) { wd[wave] = d; wj[wave] = j; }
        __syncthreads();

        if (wave == 0) {
            float dv = (lane < NWAVES) ? wd[lane] : INF_SQ;
            int   jv = (lane < NWAVES) ? wj[lane] : 0x7fffffff;
            #pragma unroll
            for (int m = 8; m >= 1; m >>= 1) {
                const float od = __shfl_xor(dv, m, 32);
                const int   oj = __shfl_xor(jv, m, 32);
                if (od < dv || (od == dv && oj < jv)) { dv = od; jv = oj; }
            }
            if (lane == 0) {
                sel_u = jv;
                const unsigned long long sj = slot[jv];
                parent_s[jv] = (int)(unsigned int)sj;        // low 32 = best_p
                slot[jv] = pack_slot(-1.0f, 0);              // mark visited
            }
        }
        __syncthreads();

        // relax: du_sq = clamped ||p_tid - p_u||^2; update unvisited (bd<0 blocks)
        const int u = sel_u;
        const float4 pu = pts[u];
        const float dx = px - pu.x, dy = py - pu.y, dz = pz - pu.z;
        const float du_sq = fmaxf(dx * dx + dy * dy + dz * dz, EPS_F);
        if (tid != u && du_sq < bd) {                        // stale bd valid: only slot[u] changed
            slot[tid] = pack_slot(du_sq, u);
        }
    }

    // ---- edge norms via WMMA Gram diagonal: cost[i] = ||pts[i]-pts[pa(i)]|| ----
    // Each wave covers rows [32*wave, 32*wave+32) as two 16x16 tiles.
    #pragma unroll
    for (int t = 0; t < 2; ++t) {
        const int r0  = wave * 32 + t * 16;
        const int row = r0 + (lane & 15);
        const int pa  = parent_s[row];
        const int pa0 = max(pa, 0);               // branchless (keeps EXEC all-ones)
        const float4 pr = pts[row];
        const float4 pp = pts[pa0];
        const float ex = pr.x - pp.x;
        const float ey = pr.y - pp.y;
        const float ez = pr.z - pp.z;
        // A (16x4 f32) layout: lanes 0-15 hold (K=0,K=1) of row M=lane,
        //                      lanes 16-31 hold (K=2,K=3) of row M=lane-16.
        // B (4x16) for Gram E*E^T has the identical per-lane image -> b = a.
        v2f a;
        a.x = (lane < 16) ? ex : ez;
        a.y = (lane < 16) ? ey : 0.0f;
        v8f c = {0.f, 0.f, 0.f, 0.f, 0.f, 0.f, 0.f, 0.f};
        c = __builtin_amdgcn_wmma_f32_16x16x4_f32(
                /*neg_a=*/false, a, /*neg_b=*/false, a,
                /*c_mod=*/(short)0, c, /*reuse_a=*/false, /*reuse_b=*/false);
        // diag(i,i): lanes 0-7 -> c[lane] (i=lane); lanes 24-31 -> c[lane-24] (i=lane-16)
        const float dsq = c[lane & 7];
        if ((lane < 8) || (lane >= 24)) {
            const float edge = sqrtf(fmaxf(dsq, EPS_F));
            cost_s[row] = (pa >= 0) ? edge : 0.0f;
        }
    }
    __syncthreads();

    // ---- mean MST edge length ----
    float s = cost_s[tid];
    #pragma unroll
    for (int m = 16; m >= 1; m >>= 1) s += __shfl_xor(s, m, 32);
    if (lane == 0) wsum[wave] = s;
    __syncthreads();
    if (tid == 0) {
        float tot = 0.0f;
        #pragma unroll
        for (int w = 0; w < NWAVES; ++w) tot += wsum[w];
        mean_sh = tot / (float)(PRIM_M - 1);
    }
    __syncthreads();

    // ---- outputs ----
    const float mean_len = mean_sh;
    const float thresh   = (*alpha_ptr) * mean_len;
    const int b = g / P;
    const int p = g - b * P;
    const size_t out_idx = (size_t)b * ((size_t)P * PRIM_M) + (size_t)p * PRIM_M + tid;
    const float cst = cost_s[tid];
    const int   par = parent_s[tid];
    const bool  mask = cst > thresh;
    dist_out[out_idx]   = mask ? cst : 0.0f;
    assign_out[out_idx] = mask ? (par + p * PRIM_M) : -1;
    if (tid == 0) mean_ws[g] = mean_len;
}

// Fixed-order reduction of per-primitive means -> mean_mst_length[B].
// (No float atomics so timed replays are bitwise deterministic.)
__global__ void mean_reduce_kernel(const float* __restrict__ mean_ws,
                                   float* __restrict__ mean_out, int B, int P)
{
    const int b = blockIdx.x * blockDim.x + threadIdx.x;
    if (b < B) {
        float s = 0.0f;
        for (int i = 0; i < P; ++i) s += mean_ws[b * P + i];
        mean_out[b] = s / (float)P;
    }
}

extern "C" void kernel_launch(void* const* d_in, const int* in_sizes, int n_in,
                              void* d_out, int out_size, void* d_ws, size_t ws_size,
                              hipStream_t stream)
{
    const float* xyz       = (const float*)d_in[0];
    // d_in[1] = primitive_size (device scalar); m = 512 fixed at compile time
    const float* alpha_ptr = (const float*)d_in[2];

    const int BN = in_sizes[0] / 3;   // B*n
    const int B  = out_size - 2 * BN; // out = dist(B*n) + assign(B*n) + mean(B)
    const int n  = BN / B;
    const int P  = n / PRIM_M;
    const int G  = BN / PRIM_M;       // B*P workgroups

    float* dist_out   = (float*)d_out;
    int*   assign_out = (int*)((float*)d_out + (size_t)BN);
    float* mean_out   = (float*)d_out + 2 * (size_t)BN;
    float* mean_ws    = (float*)d_ws;

    mst_expansion_kernel<<<G, PRIM_M, 0, stream>>>(
        xyz, alpha_ptr, dist_out, assign_out, mean_ws, P);
    mean_reduce_kernel<<<(B + 63) / 64, 64, 0, stream>>>(mean_ws, mean_out, B, P);
}